// InitReduceConv_38680475468439
// MI455X (gfx1250) — compile-verified
//
#include <hip/hip_runtime.h>

// segment_sum(boundary_x[src], dst, 100000) :
//   gather 128-f32 rows, scatter-add into output rows.
// Memory-bound: x (51.2MB) and out (51.2MB) both fit in the 192MB L2 of
// MI455X, so after the cold pass all gather/atomic traffic is L2-resident.
// One wave32 per edge: lane l loads float4 x[src][4l..4l+3] (global_load_b128,
// 512B coalesced per wave) and issues 4 non-returning device-scope
// global_atomic_add_f32 into out[dst][4l..4l+3].

#define D_FEAT 128

__global__ void __launch_bounds__(256)
zero_out_kernel(float4* __restrict__ out, int n4) {
    int i = blockIdx.x * blockDim.x + threadIdx.x;
    if (i < n4) out[i] = make_float4(0.f, 0.f, 0.f, 0.f);
}

// Non-returning HW float atomic with immediate offset, device scope.
// Inline asm guarantees global_atomic_add_f32 (no CAS fallback) and lets the
// compiler keep one 64-bit address pair for all four adds (offsets folded).
template <int OFF>
__device__ __forceinline__ void gatomic_add_f32(float* p, float v) {
    asm volatile("global_atomic_add_f32 %0, %1, off offset:%2 scope:SCOPE_DEV"
                 :
                 : "v"(p), "v"(v), "i"(OFF)
                 : "memory");
}

__global__ void __launch_bounds__(256)
gather_scatter_add_kernel(const float* __restrict__ x,
                          const int* __restrict__ idx_src,
                          const int* __restrict__ idx_dst,
                          float* __restrict__ out,
                          int n_edges) {
    const int gid  = blockIdx.x * blockDim.x + threadIdx.x;
    const int e    = gid >> 5;          // one wave32 per edge
    const int lane = gid & 31;
    if (e >= n_edges) return;

    const int s = idx_src[e];           // uniform within wave; one cacheline
    const int d = idx_dst[e];

    // 32 lanes x float4 = full 512B row, coalesced -> global_load_b128
    const float4 v =
        ((const float4*)(x + (size_t)s * D_FEAT))[lane];

    float* o = out + (size_t)d * D_FEAT + (lane << 2);
    gatomic_add_f32<0>(o, v.x);
    gatomic_add_f32<4>(o, v.y);
    gatomic_add_f32<8>(o, v.z);
    gatomic_add_f32<12>(o, v.w);
    // s_endpgm implies wait-idle: all atomics retire before wave exit.
}

extern "C" void kernel_launch(void* const* d_in, const int* in_sizes, int n_in,
                              void* d_out, int out_size, void* d_ws, size_t ws_size,
                              hipStream_t stream) {
    const float* x      = (const float*)d_in[0];
    const int*   bindex = (const int*)d_in[1];      // [2, E] int32, row-major
    const int    E      = in_sizes[1] / 2;
    float*       out    = (float*)d_out;

    // 1) zero the output (harness poisons it)
    const int n4 = out_size / 4;                    // 128 is divisible by 4
    zero_out_kernel<<<(n4 + 255) / 256, 256, 0, stream>>>((float4*)out, n4);

    // 2) gather + scatter-add, one wave per edge
    const long long threads = (long long)E * 32;
    const int blocks = (int)((threads + 255) / 256);
    gather_scatter_add_kernel<<<blocks, 256, 0, stream>>>(
        x, bindex, bindex + E, out, E);
}